// GraphNorm_88536455840506
// MI455X (gfx1250) — compile-verified
//
#include <hip/hip_runtime.h>
#include <hip/hip_bf16.h>

// GraphNorm for MI455X (gfx1250).
// Memory-bound (~1.5 GB traffic -> ~66us @ 23.3 TB/s). Two passes:
//   Pass 1 (gn_stats): per-segment sum / sum-of-squares via WMMA reduction of
//           LDS tiles staged with CDNA5 async global->LDS copies. No atomics
//           (batch is sorted -> binary-searched contiguous runs).
//   Pass 2 (gn_apply): fused affine normalize, float4 streaming.
//
// Assumes D == 128 (reference setup), wave32.

typedef __attribute__((ext_vector_type(2))) float v2f;
typedef __attribute__((ext_vector_type(8))) float v8f;

#define D_FEAT 128
#define TILE_ROWS 32
#define GN_EPS 1e-8f

__global__ void __launch_bounds__(256) gn_stats(
    const float* __restrict__ x, const int* __restrict__ batch,
    const int* __restrict__ Gptr, int N, float* __restrict__ ws)
{
    __shared__ float tile[TILE_ROWS * D_FEAT];   // 16 KB staging tile

    const int G = Gptr[0];
    float* __restrict__ meanBuf = ws;                          // [G, 128]
    float* __restrict__ rstdBuf = ws + (size_t)G * D_FEAT;     // [G, 128]

    const int tid   = threadIdx.x;
    const int lane  = tid & 31;
    const int wave  = tid >> 5;           // 0..7, each wave owns 16 columns
    const int c0    = wave * 16;
    const int ncol  = lane & 15;          // N index within wave's 16 columns
    const int khalf = (lane >> 4) * 2;    // lanes 16-31 carry the other K half

    v2f ones; ones[0] = 1.0f; ones[1] = 1.0f;   // A matrix: 16x4 all-ones

    for (int g = blockIdx.x; g < G; g += gridDim.x) {
        // --- binary search contiguous segment [start, end) in sorted batch ---
        int lo = 0, hi = N;
        while (lo < hi) { int mid = (lo + hi) >> 1; if (batch[mid] < g) lo = mid + 1; else hi = mid; }
        const int start = lo;
        lo = start; hi = N;
        while (lo < hi) { int mid = (lo + hi) >> 1; if (batch[mid] < g + 1) lo = mid + 1; else hi = mid; }
        const int end = lo;
        const int cnt = end - start;

        v8f acc1 = {};   // column sums        (16x16 f32 C/D, all rows equal)
        v8f acc2 = {};   // column sum-squares

        for (int r0 = start; r0 < end; r0 += TILE_ROWS) {
            const int valid = min(end - r0, TILE_ROWS);

            // ---- stage tile: async global->LDS b128, zero-pad tail rows ----
            for (int q = tid; q < TILE_ROWS * (D_FEAT / 4); q += 256) {
                const int row = q >> 5;                 // 32 float4 per row
                if (row < valid) {
                    const float* src = x + (size_t)(r0 + row) * D_FEAT + (q & 31) * 4;
                    unsigned lds_off = (unsigned)(q * 16);  // tile is at LDS offset 0
                    asm volatile("global_load_async_to_lds_b128 %0, %1, off"
                                 :: "v"(lds_off), "v"(src) : "memory");
                } else {
                    float4 z = make_float4(0.f, 0.f, 0.f, 0.f);
                    *(float4*)&tile[q * 4] = z;
                }
            }
            asm volatile("s_wait_asynccnt 0" ::: "memory");
            __syncthreads();

            // ---- WMMA reduction: 8 chunks of 4 rows; D += ones(16x4) x B(4x16)
            // A all-ones => result invariant to K-layout of B; only the N<->lane
            // mapping (matches documented C/D layout) matters.
#pragma unroll
            for (int kc = 0; kc < TILE_ROWS; kc += 4) {
                const float e0 = tile[(kc + khalf + 0) * D_FEAT + c0 + ncol];
                const float e1 = tile[(kc + khalf + 1) * D_FEAT + c0 + ncol];
                v2f b1, b2;
                b1[0] = e0;      b1[1] = e1;
                b2[0] = e0 * e0; b2[1] = e1 * e1;
                acc1 = __builtin_amdgcn_wmma_f32_16x16x4_f32(
                           false, ones, false, b1, (short)0, acc1, false, false);
                acc2 = __builtin_amdgcn_wmma_f32_16x16x4_f32(
                           false, ones, false, b2, (short)0, acc2, false, false);
            }
            __syncthreads();   // protect LDS before next tile is staged
        }

        // ---- finalize: row M=0 of D lives in VGPR0 of lanes 0..15 ----
        if (lane < 16) {
            const float s1   = acc1[0];
            const float s2   = acc2[0];
            const float cntf = (float)cnt;
            const float cmax = fmaxf(cntf, 1.0f);
            const float mean = s1 / cmax;
            const float var  = (s2 - cmax * mean * mean) / fmaxf(cntf - 1.0f, 1.0f);
            const float rstd = 1.0f / (sqrtf(fmaxf(var, 0.0f)) + GN_EPS);
            meanBuf[(size_t)g * D_FEAT + c0 + ncol] = mean;
            rstdBuf[(size_t)g * D_FEAT + c0 + ncol] = rstd;
        }
    }
}

__global__ void __launch_bounds__(256) gn_apply(
    const float* __restrict__ x, const int* __restrict__ batch,
    const float* __restrict__ weight, const float* __restrict__ bias,
    const float* __restrict__ mscale, const int* __restrict__ Gptr,
    int N, float* __restrict__ out, const float* __restrict__ ws)
{
    const int G = Gptr[0];
    const float4* __restrict__ X4 = (const float4*)x;
    float4*       __restrict__ O4 = (float4*)out;
    const float4* __restrict__ M4 = (const float4*)ws;
    const float4* __restrict__ R4 = (const float4*)(ws + (size_t)G * D_FEAT);
    const float4* __restrict__ W4 = (const float4*)weight;
    const float4* __restrict__ B4 = (const float4*)bias;
    const float4* __restrict__ S4 = (const float4*)mscale;

    const size_t total4 = (size_t)N * (D_FEAT / 4);
    const size_t stride = (size_t)gridDim.x * blockDim.x;

    for (size_t i = (size_t)blockIdx.x * blockDim.x + threadIdx.x;
         i < total4; i += stride) {
        const size_t row = i >> 5;          // 32 float4 per row
        const int    c4  = (int)(i & 31);
        const int    g   = batch[row];      // sorted -> L2-resident stat rows

        __builtin_prefetch(&X4[i + stride], 0, 1);   // global_prefetch_b8

        const float4 xv = X4[i];
        const float4 mv = M4[(size_t)g * 32 + c4];
        const float4 rv = R4[(size_t)g * 32 + c4];
        const float4 wv = W4[c4];
        const float4 bv = B4[c4];
        const float4 sv = S4[c4];

        float4 o;
        o.x = wv.x * (xv.x - mv.x * sv.x) * rv.x + bv.x;
        o.y = wv.y * (xv.y - mv.y * sv.y) * rv.y + bv.y;
        o.z = wv.z * (xv.z - mv.z * sv.z) * rv.z + bv.z;
        o.w = wv.w * (xv.w - mv.w * sv.w) * rv.w + bv.w;
        O4[i] = o;
    }
}

extern "C" void kernel_launch(void* const* d_in, const int* in_sizes, int n_in,
                              void* d_out, int out_size, void* d_ws, size_t ws_size,
                              hipStream_t stream) {
    const float* x      = (const float*)d_in[0];
    const int*   batch  = (const int*)d_in[1];
    const float* weight = (const float*)d_in[2];
    const float* bias   = (const float*)d_in[3];
    const float* mscale = (const float*)d_in[4];
    const int*   Gptr   = (const int*)d_in[5];

    const int D = in_sizes[2];            // 128 per reference setup
    const int N = in_sizes[0] / D;        // 1,000,000
    float* out = (float*)d_out;
    float* ws  = (float*)d_ws;            // mean[G*D] then rstd[G*D]

    // Pass 1: per-segment stats (grid-stride over G read on-device).
    gn_stats<<<8192, 256, 0, stream>>>(x, batch, Gptr, N, ws);

    // Pass 2: fused normalize, ~4 float4 per thread.
    const size_t total4 = (size_t)N * (D / 4);
    int blocks = (int)((total4 + 1023) / 1024);
    if (blocks < 1) blocks = 1;
    gn_apply<<<blocks, 256, 0, stream>>>(x, batch, weight, bias, mscale,
                                         Gptr, N, out, ws);
}